// FastWeightLayer_82652350644603
// MI455X (gfx1250) — compile-verified
//
#include <hip/hip_runtime.h>
#include <hip/hip_bf16.h>
#include <math.h>

typedef __attribute__((ext_vector_type(2))) float v2f;
typedef __attribute__((ext_vector_type(8))) float v8f;

#define T_DIM 256
#define H_DIM 512
#define LN_EPS 1e-5f

// ---------------- WMMA helper: D = A(16x4,f32) x B(4x16,f32) + C ----------------
__device__ __forceinline__ v8f wmma4(v2f a, v2f b, v8f c) {
    // 8-arg form: (neg_a, A, neg_b, B, c_mod, C, reuse_a, reuse_b)
    return __builtin_amdgcn_wmma_f32_16x16x4_f32(false, a, false, b, (short)0, c,
                                                 false, false);
}

// One wave computes a 16x16 f32 tile of A(row-major, lda) @ B(row-major, ldb).
__device__ __forceinline__ v8f tile_mm(const float* __restrict__ A, int lda,
                                       const float* __restrict__ B, int ldb,
                                       int tm, int tn, int K) {
    const int lane = threadIdx.x & 31;
    const int rw   = lane & 15;          // M index for A loads, N index for B loads
    const int kh   = (lane >> 4) << 1;   // 0 or 2 (K sub-offset per lane half)
    const float* ap = A + (tm + rw) * lda + kh;
    const float* bp = B + kh * ldb + (tn + rw);
    v8f acc = {};
    for (int k = 0; k < K; k += 4) {
        v2f av, bv;
        av.x = ap[0];  av.y = ap[1];
        bv.x = bp[0];  bv.y = bp[ldb];
        acc = wmma4(av, bv, acc);
        ap += 4;
        bp += 4 * ldb;
    }
    return acc;
}

__device__ __forceinline__ void tile_store(float* __restrict__ C, int ldc,
                                           int tm, int tn, v8f acc) {
    const int lane  = threadIdx.x & 31;
    const int col   = lane & 15;
    const int rbase = (lane >> 4) * 8;
    float* cp = C + (tm + rbase) * ldc + tn + col;
#pragma unroll
    for (int r = 0; r < 8; ++r) cp[r * ldc] = acc[r];
}

// ---------------- Kernel 1: Hcum[t][q] = inclusive cumsum_t h ----------------
__global__ void k_colcum(const float* __restrict__ h, float* __restrict__ Hcum) {
    int q = blockIdx.x * blockDim.x + threadIdx.x;
    if (q >= H_DIM) return;
    float s = 0.f;
    for (int t = 0; t < T_DIM; ++t) {
        s += h[t * H_DIM + q];
        Hcum[t * H_DIM + q] = s;
    }
}

// ---------------- Kernel 2: c[t] = sum_q h[t][q] (one wave per row) ----------------
__global__ void k_rowsum(const float* __restrict__ h, float* __restrict__ c) {
    int wid  = (blockIdx.x * blockDim.x + threadIdx.x) >> 5;
    int lane = threadIdx.x & 31;
    if (wid >= T_DIM) return;
    float s = 0.f;
    for (int q = lane; q < H_DIM; q += 32) s += h[wid * H_DIM + q];
#pragma unroll
    for (int off = 16; off > 0; off >>= 1) s += __shfl_down(s, off, 32);
    if (lane == 0) c[wid] = s;
}

// ---------------- Kernel 3: z = relu(h@U + a); u = h*z ----------------
__global__ __launch_bounds__(256) void k_gemm_relu_u(
    const float* __restrict__ h, const float* __restrict__ U,
    const float* __restrict__ a, float* __restrict__ z, float* __restrict__ u) {
    int wave = (blockIdx.x * blockDim.x + threadIdx.x) >> 5;
    int tm = (wave >> 5) << 4;   // 32 tiles along N
    int tn = (wave & 31) << 4;
    v8f acc = tile_mm(h, H_DIM, U, H_DIM, tm, tn, H_DIM);
    const int lane = threadIdx.x & 31;
    const int col = lane & 15, rbase = (lane >> 4) * 8;
#pragma unroll
    for (int r = 0; r < 8; ++r) {
        int row = tm + rbase + r, cc = tn + col;
        float zi = fmaxf(acc[r] + a[cc], 0.f);
        z[row * H_DIM + cc] = zi;
        u[row * H_DIM + cc] = zi * h[row * H_DIM + cc];
    }
}

// ---------------- Kernel 4: p = z@W + b ----------------
__global__ __launch_bounds__(256) void k_gemm_bias(
    const float* __restrict__ z, const float* __restrict__ W,
    const float* __restrict__ b, float* __restrict__ p) {
    int wave = (blockIdx.x * blockDim.x + threadIdx.x) >> 5;
    int tm = (wave >> 5) << 4;
    int tn = (wave & 31) << 4;
    v8f acc = tile_mm(z, H_DIM, W, H_DIM, tm, tn, H_DIM);
    const int lane = threadIdx.x & 31;
    const int col = lane & 15, rbase = (lane >> 4) * 8;
#pragma unroll
    for (int r = 0; r < 8; ++r) {
        int row = tm + rbase + r, cc = tn + col;
        p[row * H_DIM + cc] = acc[r] + b[cc];
    }
}

// ---------------- block reductions ----------------
__device__ __forceinline__ float block_sum(float v, float* red) {
    int tid = threadIdx.x;
    red[tid] = v; __syncthreads();
    for (int s = 128; s > 0; s >>= 1) {
        if (tid < s) red[tid] += red[tid + s];
        __syncthreads();
    }
    float r = red[0]; __syncthreads();
    return r;
}
__device__ __forceinline__ float block_max(float v, float* red) {
    int tid = threadIdx.x;
    red[tid] = v; __syncthreads();
    for (int s = 128; s > 0; s >>= 1) {
        if (tid < s) red[tid] = fmaxf(red[tid], red[tid + s]);
        __syncthreads();
    }
    float r = red[0]; __syncthreads();
    return r;
}

// ---------------- Kernel 5: per-row CE(LN(p)) backward -> d, dh = d*Hcum ----------------
__global__ __launch_bounds__(256) void k_ln_grad(
    const float* __restrict__ p, const float* __restrict__ gamma,
    const float* __restrict__ beta, const int* __restrict__ targets,
    const float* __restrict__ Hcum, float* __restrict__ d, float* __restrict__ dh) {
    __shared__ float red[256];
    const int t = blockIdx.x, tid = threadIdx.x;
    const int i0 = t * H_DIM + tid, i1 = i0 + 256;
    float x0 = p[i0], x1 = p[i1];
    float m = block_sum(x0 + x1, red) * (1.f / H_DIM);
    float a0 = x0 - m, a1 = x1 - m;
    float var = block_sum(a0 * a0 + a1 * a1, red) * (1.f / H_DIM);
    float r = rsqrtf(var + LN_EPS);
    float h0 = a0 * r, h1 = a1 * r;                 // normalized
    float g0 = gamma[tid], g1 = gamma[tid + 256];
    float y0 = h0 * g0 + beta[tid], y1 = h1 * g1 + beta[tid + 256];
    float mx = block_max(fmaxf(y0, y1), red);
    float e0 = expf(y0 - mx), e1 = expf(y1 - mx);
    float se = block_sum(e0 + e1, red);
    int tgt = targets[t];
    float dy0 = e0 / se - (tid == tgt ? 1.f : 0.f);
    float dy1 = e1 / se - (tid + 256 == tgt ? 1.f : 0.f);
    float dg0 = dy0 * g0, dg1 = dy1 * g1;           // dL/d(normalized)
    float s1 = block_sum(dg0 + dg1, red) * (1.f / H_DIM);
    float s2 = block_sum(dg0 * h0 + dg1 * h1, red) * (1.f / H_DIM);
    float d0 = r * (dg0 - s1 - h0 * s2);
    float d1 = r * (dg1 - s1 - h1 * s2);
    d[i0] = d0;  d[i1] = d1;
    dh[i0] = d0 * Hcum[i0];  dh[i1] = d1 * Hcum[i1];
}

// ---------------- Kernel 6: Bm[t][i] = (z_t . u_i + c_i) * [i<t] ----------------
__global__ __launch_bounds__(256) void k_alpha(
    const float* __restrict__ z, const float* __restrict__ u,
    const float* __restrict__ c, float* __restrict__ Bm) {
    int wave = (blockIdx.x * blockDim.x + threadIdx.x) >> 5;
    int tm = (wave >> 4) << 4;   // 16 tiles along N (=T)
    int tn = (wave & 15) << 4;
    const int lane = threadIdx.x & 31;
    const int rw = lane & 15;
    const int kh = (lane >> 4) << 1;
    // A = z (row-major); B = u^T so B[k][n] = u[n][k] -> A-style loads from u
    const float* ap = z + (tm + rw) * H_DIM + kh;
    const float* bp = u + (tn + rw) * H_DIM + kh;
    v8f acc = {};
    for (int k = 0; k < H_DIM; k += 4) {
        v2f av, bv;
        av.x = ap[0]; av.y = ap[1];
        bv.x = bp[0]; bv.y = bp[1];
        acc = wmma4(av, bv, acc);
        ap += 4; bp += 4;
    }
    const int col = lane & 15, rbase = (lane >> 4) * 8;
#pragma unroll
    for (int r = 0; r < 8; ++r) {
        int tt = tm + rbase + r, ii = tn + col;
        Bm[tt * T_DIM + ii] = (ii < tt) ? (acc[r] + c[ii]) : 0.f;
    }
}

// ---------------- Kernel 7: P1 = Bm@D, P2 = Bm@Dh (shared A loads) ----------------
__global__ __launch_bounds__(256) void k_pgemm(
    const float* __restrict__ Bm, const float* __restrict__ D,
    const float* __restrict__ Dh, float* __restrict__ P1, float* __restrict__ P2) {
    int wave = (blockIdx.x * blockDim.x + threadIdx.x) >> 5;
    int tm = (wave >> 5) << 4;
    int tn = (wave & 31) << 4;
    const int lane = threadIdx.x & 31;
    const int rw = lane & 15;
    const int kh = (lane >> 4) << 1;
    const float* ap = Bm + (tm + rw) * T_DIM + kh;
    const float* b1 = D + kh * H_DIM + tn + rw;
    const float* b2 = Dh + kh * H_DIM + tn + rw;
    v8f acc1 = {}, acc2 = {};
    for (int k = 0; k < T_DIM; k += 4) {
        v2f av, bv1, bv2;
        av.x = ap[0]; av.y = ap[1];
        bv1.x = b1[0]; bv1.y = b1[H_DIM];
        bv2.x = b2[0]; bv2.y = b2[H_DIM];
        acc1 = wmma4(av, bv1, acc1);
        acc2 = wmma4(av, bv2, acc2);
        ap += 4; b1 += 4 * H_DIM; b2 += 4 * H_DIM;
    }
    tile_store(P1, H_DIM, tm, tn, acc1);
    tile_store(P2, H_DIM, tm, tn, acc2);
}

// ---------------- Kernel 8: pre = p - (Hcum*P1 - P2); out = LN(pre) ----------------
__global__ __launch_bounds__(256) void k_final(
    const float* __restrict__ p, const float* __restrict__ Hcum,
    const float* __restrict__ P1, const float* __restrict__ P2,
    const float* __restrict__ gamma, const float* __restrict__ beta,
    float* __restrict__ out) {
    __shared__ float red[256];
    const int t = blockIdx.x, tid = threadIdx.x;
    const int i0 = t * H_DIM + tid, i1 = i0 + 256;
    float pre0 = p[i0] - (Hcum[i0] * P1[i0] - P2[i0]);
    float pre1 = p[i1] - (Hcum[i1] * P1[i1] - P2[i1]);
    float m = block_sum(pre0 + pre1, red) * (1.f / H_DIM);
    float a0 = pre0 - m, a1 = pre1 - m;
    float var = block_sum(a0 * a0 + a1 * a1, red) * (1.f / H_DIM);
    float r = rsqrtf(var + LN_EPS);
    out[i0] = a0 * r * gamma[tid] + beta[tid];
    out[i1] = a1 * r * gamma[tid + 256] + beta[tid + 256];
}

extern "C" void kernel_launch(void* const* d_in, const int* in_sizes, int n_in,
                              void* d_out, int out_size, void* d_ws, size_t ws_size,
                              hipStream_t stream) {
    const float* h     = (const float*)d_in[0];   // (1,T,H)
    const float* U     = (const float*)d_in[1];   // (H,H)
    const float* W     = (const float*)d_in[2];   // (H,H)
    const float* a     = (const float*)d_in[3];   // (H)
    const float* b     = (const float*)d_in[4];   // (H)
    const float* gamma = (const float*)d_in[5];   // (H)
    const float* beta  = (const float*)d_in[6];   // (H)
    const int*   tgt   = (const int*)d_in[7];     // (T)
    float* out = (float*)d_out;                   // (1,T,H)

    float* ws = (float*)d_ws;
    const size_t TH = (size_t)T_DIM * H_DIM;      // 131072
    float* z    = ws;                // T*H
    float* u    = ws + 1 * TH;       // T*H
    float* p    = ws + 2 * TH;       // T*H
    float* Hcum = ws + 3 * TH;       // T*H
    float* d    = ws + 4 * TH;       // T*H
    float* dh   = ws + 5 * TH;       // T*H
    float* P1   = ws + 6 * TH;       // T*H
    float* P2   = ws + 7 * TH;       // T*H
    float* Bm   = ws + 8 * TH;       // T*T = 65536
    float* c    = Bm + (size_t)T_DIM * T_DIM;  // T

    // prefix sums / row sums of h
    k_colcum<<<H_DIM / 256, 256, 0, stream>>>(h, Hcum);
    k_rowsum<<<T_DIM / 8, 256, 0, stream>>>(h, c);
    // z = relu(h@U + a), u = h*z   (512 tiles, 8 waves/block)
    k_gemm_relu_u<<<64, 256, 0, stream>>>(h, U, a, z, u);
    // p = z@W + b
    k_gemm_bias<<<64, 256, 0, stream>>>(z, W, b, p);
    // per-row LN+CE backward -> d, dh
    k_ln_grad<<<T_DIM, 256, 0, stream>>>(p, gamma, beta, tgt, Hcum, d, dh);
    // Bm = mask(z@u^T + c)   (256 tiles)
    k_alpha<<<32, 256, 0, stream>>>(z, u, c, Bm);
    // P1 = Bm@d, P2 = Bm@dh
    k_pgemm<<<64, 256, 0, stream>>>(Bm, d, dh, P1, P2);
    // pre = p - (Hcum*P1 - P2); out = LN(pre)
    k_final<<<T_DIM, 256, 0, stream>>>(p, Hcum, P1, P2, gamma, beta, out);
}